// LinearSelfAttention_15118284882355
// MI455X (gfx1250) — compile-verified
//
#include <hip/hip_runtime.h>
#include <hip/hip_bf16.h>
#include <math.h>

// LinearSelfAttention (MobileViT) for MI455X / gfx1250.
// Rewrite: ctx = (softmax(q)^T x) @ W_k + b_k   (since sum(scores)==1)
// so the only heavy compute is:  out = (relu(x@W_v + b_v) * ctx) @ W_o + b_o
// -> two chained 131072x256x256 fp32 GEMMs, done with V_WMMA_F32_16X16X4_F32.

#define EMB   256
#define NTOK  1024
#define NBP   128            // B*P
#define MROWS (NBP * NTOK)   // 131072
#define EQKV  513

typedef __attribute__((ext_vector_type(2))) float v2f;
typedef __attribute__((ext_vector_type(8))) float v8f;

#if defined(__has_builtin)
#if __has_builtin(__builtin_amdgcn_global_load_async_to_lds_b64) && \
    __has_builtin(__builtin_amdgcn_s_wait_asynccnt)
#define HAVE_ASYNC_LDS 1
#endif
#endif

#ifdef HAVE_ASYNC_LDS
// Exact parameter types per hipcc diagnostic: v2i in AS1 (global), AS3 (lds).
typedef int v2i __attribute__((__vector_size__(8)));
typedef __attribute__((address_space(1))) v2i gv2i;
typedef __attribute__((address_space(3))) v2i lv2i;
#endif

__device__ __forceinline__ v8f wmma_f32x4(v2f a, v2f b, v8f c) {
  // (neg_a, A, neg_b, B, c_mod, C, reuse_a, reuse_b)
  return __builtin_amdgcn_wmma_f32_16x16x4_f32(false, a, false, b, (short)0, c,
                                               false, false);
}

// ---------------------------------------------------------------------------
// Kernel A: q[m] = x[m,:] . W_qkv[:,0] + b_qkv[0]    (one wave per row)
// ---------------------------------------------------------------------------
__global__ __launch_bounds__(256) void q_gemv_kernel(
    const float* __restrict__ x, const float* __restrict__ Wqkv,
    const float* __restrict__ bqkv, float* __restrict__ q) {
  const int lane = threadIdx.x & 31;
  const int wid  = threadIdx.x >> 5;
  const int row  = blockIdx.x * 8 + wid;
  const float* xr = x + (size_t)row * EMB;
  float s = 0.f;
#pragma unroll
  for (int i = 0; i < 8; ++i) {
    const int k = lane + 32 * i;
    s += xr[k] * Wqkv[(size_t)k * EQKV];   // column 0 of W_qkv
  }
#pragma unroll
  for (int off = 16; off > 0; off >>= 1) s += __shfl_xor(s, off, 32);
  if (lane == 0) q[row] = s + bqkv[0];
}

// ---------------------------------------------------------------------------
// Kernel B: per (b,p): softmax over q[1024], xbar = sum_n s_n * x[n,:],
//           ctx = xbar @ W_k + b_k.      One 256-thread WG per (b,p).
// ---------------------------------------------------------------------------
__global__ __launch_bounds__(256) void ctx_kernel(
    const float* __restrict__ x, const float* __restrict__ Wqkv,
    const float* __restrict__ bqkv, const float* __restrict__ q,
    float* __restrict__ ctx) {
  __shared__ float qs[NTOK];
  __shared__ float red[256];
  __shared__ float xb[EMB];
  const int t  = threadIdx.x;
  const int bp = blockIdx.x;

  const float* qsrc = q + (size_t)bp * NTOK;
  float lmax = -INFINITY;
#pragma unroll
  for (int i = 0; i < 4; ++i) {
    const float v = qsrc[t + 256 * i];
    qs[t + 256 * i] = v;
    lmax = fmaxf(lmax, v);
  }
  red[t] = lmax;
  __syncthreads();
  for (int s = 128; s > 0; s >>= 1) {
    if (t < s) red[t] = fmaxf(red[t], red[t + s]);
    __syncthreads();
  }
  const float mx = red[0];
  __syncthreads();

  float lsum = 0.f;
#pragma unroll
  for (int i = 0; i < 4; ++i) {
    const float e = __expf(qs[t + 256 * i] - mx);
    qs[t + 256 * i] = e;
    lsum += e;
  }
  red[t] = lsum;
  __syncthreads();
  for (int s = 128; s > 0; s >>= 1) {
    if (t < s) red[t] += red[t + s];
    __syncthreads();
  }
  const float inv = 1.f / red[0];
  __syncthreads();

  // xbar[t] = (sum_n e_n * x[n,t]) * inv   -- coalesced over t
  const float* xcol = x + (size_t)bp * NTOK * EMB + t;
  float acc = 0.f;
#pragma unroll 4
  for (int n = 0; n < NTOK; ++n) acc += qs[n] * xcol[(size_t)n * EMB];
  xb[t] = acc * inv;
  __syncthreads();

  // ctx[t] = xbar . W_qkv[:, 1+t] + b_qkv[1+t]
  float acc2 = 0.f;
#pragma unroll 4
  for (int k = 0; k < EMB; ++k) acc2 += xb[k] * Wqkv[(size_t)k * EQKV + 1 + t];
  ctx[(size_t)bp * EMB + t] = acc2 + bqkv[1 + t];
}

// ---------------------------------------------------------------------------
// Kernel C: out = (relu(x @ W_v + b_v) * ctx) @ W_o + b_o
// 256 threads = 8 waves; 32-row tile; WMMA f32 16x16x4.
// Wave w owns the full 32 M rows x 32 N cols (slab n0w = w*32): 2x2 sub-tiles,
// so every A/B fragment feeds two wmma.
// B stage is kept in fragment-ready interleaved layout:
//   BsP[kp][n] = (B[2kp][n], B[2kp+1][n])  -> one conflict-free ds_load_b64.
// ---------------------------------------------------------------------------
#define BM         32
#define KSTEP      16
#define U_STRIDE   260   // 260 % 64 == 4  -> conflict-free phase-2 A-frags
#define AS_STRIDE  20    // groups of 4 words tile all 64 banks
#define BSP_STRIDE 272   // float2 units; 544 % 64 == 32 -> halves disjoint

__global__ __launch_bounds__(256, 2) void fused_vout_kernel(
    const float* __restrict__ x, const float* __restrict__ Wqkv,
    const float* __restrict__ bqkv, const float* __restrict__ Wo,
    const float* __restrict__ bo, const float* __restrict__ ctx,
    float* __restrict__ out) {
  __shared__ float U[BM * U_STRIDE];                  // 33280 B
  __shared__ float As[BM * AS_STRIDE];                //  2560 B
  __shared__ v2f   BsP[(KSTEP / 2) * BSP_STRIDE];     // 17408 B

  const int t    = threadIdx.x;
  const int lane = t & 31;
  const int wid  = t >> 5;
  const int half = lane >> 4;
  const int l16  = lane & 15;
  const int n0w  = wid * 32;     // wave's 32-col slab
  const int row0 = blockIdx.x * BM;
  const int bp   = row0 >> 10;   // 1024 tokens per (b,p); BM divides 1024

  // A-stage addressing for this thread (one float2 per thread per stage)
  const int ar = t >> 3;
  const int ac = (t & 7) * 2;
  const float* agp_base = x + (size_t)(row0 + ar) * EMB + ac;
  float* alp = &As[ar * AS_STRIDE + ac];

  const v8f vzero = {};
  v8f acc[2][2];
#pragma unroll
  for (int i = 0; i < 2; ++i)
#pragma unroll
    for (int j = 0; j < 2; ++j) acc[i][j] = vzero;

  // ---------------- Phase 1: U = relu(X @ W_v + b_v) * ctx ----------------
  for (int k0 = 0; k0 < EMB; k0 += KSTEP) {
    // stage A tile: BM x 16 of x  (async HBM -> LDS when available)
#ifdef HAVE_ASYNC_LDS
    {
      float* gsrc = const_cast<float*>(agp_base + k0);
      __builtin_amdgcn_global_load_async_to_lds_b64((gv2i*)gsrc, (lv2i*)alp, 0,
                                                    0);
    }
#else
    *(v2f*)alp = *(const v2f*)(agp_base + k0);
#endif
    // stage B tile: 16 x 256 of W_v = W_qkv[:, 257:513], interleaved pairs
#pragma unroll
    for (int i = 0; i < KSTEP / 2; ++i) {
      v2f bv;
      bv.x = Wqkv[(size_t)(k0 + 2 * i) * EQKV + 257 + t];
      bv.y = Wqkv[(size_t)(k0 + 2 * i + 1) * EQKV + 257 + t];
      BsP[i * BSP_STRIDE + t] = bv;
    }
#ifdef HAVE_ASYNC_LDS
    __builtin_amdgcn_s_wait_asynccnt(0);
#endif
    __syncthreads();

#pragma unroll
    for (int kk = 0; kk < KSTEP; kk += 4) {
      const int kc = kk + 2 * half;
      const v2f a0 = *(const v2f*)&As[(0 + l16) * AS_STRIDE + kc];
      const v2f a1 = *(const v2f*)&As[(16 + l16) * AS_STRIDE + kc];
      const int kp = (kk >> 1) + half;
      const v2f b0 = BsP[kp * BSP_STRIDE + n0w + l16];
      const v2f b1 = BsP[kp * BSP_STRIDE + n0w + 16 + l16];
      acc[0][0] = wmma_f32x4(a0, b0, acc[0][0]);
      acc[0][1] = wmma_f32x4(a0, b1, acc[0][1]);
      acc[1][0] = wmma_f32x4(a1, b0, acc[1][0]);
      acc[1][1] = wmma_f32x4(a1, b1, acc[1][1]);
    }
    __syncthreads();
  }

  // epilogue 1: bias, relu, scale by ctx; park U in LDS; reset accumulators
#pragma unroll
  for (int i = 0; i < 2; ++i)
#pragma unroll
    for (int j = 0; j < 2; ++j) {
      const int col = n0w + j * 16 + l16;
      const float bv = bqkv[257 + col];
      const float cv = ctx[(size_t)bp * EMB + col];
#pragma unroll
      for (int r = 0; r < 8; ++r) {
        const int m = i * 16 + r + 8 * half;
        float u = acc[i][j][r] + bv;
        u = u > 0.f ? u : 0.f;
        U[m * U_STRIDE + col] = u * cv;
      }
      acc[i][j] = vzero;
    }
  __syncthreads();

  // ---------------- Phase 2: OUT = U @ W_o + b_o ----------------
  for (int k0 = 0; k0 < EMB; k0 += KSTEP) {
    // stage B tile: 16 x 256 of W_o, interleaved pairs
#pragma unroll
    for (int i = 0; i < KSTEP / 2; ++i) {
      v2f bv;
      bv.x = Wo[(size_t)(k0 + 2 * i) * EMB + t];
      bv.y = Wo[(size_t)(k0 + 2 * i + 1) * EMB + t];
      BsP[i * BSP_STRIDE + t] = bv;
    }
    __syncthreads();

#pragma unroll
    for (int kk = 0; kk < KSTEP; kk += 4) {
      const int kc = k0 + kk + 2 * half;
      const v2f a0 = *(const v2f*)&U[(0 + l16) * U_STRIDE + kc];
      const v2f a1 = *(const v2f*)&U[(16 + l16) * U_STRIDE + kc];
      const int kp = (kk >> 1) + half;
      const v2f b0 = BsP[kp * BSP_STRIDE + n0w + l16];
      const v2f b1 = BsP[kp * BSP_STRIDE + n0w + 16 + l16];
      acc[0][0] = wmma_f32x4(a0, b0, acc[0][0]);
      acc[0][1] = wmma_f32x4(a0, b1, acc[0][1]);
      acc[1][0] = wmma_f32x4(a1, b0, acc[1][0]);
      acc[1][1] = wmma_f32x4(a1, b1, acc[1][1]);
    }
    __syncthreads();
  }

  // epilogue 2: add b_o, write out (16-lane segments write 64B contiguous)
#pragma unroll
  for (int i = 0; i < 2; ++i)
#pragma unroll
    for (int j = 0; j < 2; ++j) {
      const int col = n0w + j * 16 + l16;
      const float bb = bo[col];
#pragma unroll
      for (int r = 0; r < 8; ++r) {
        const int m = row0 + i * 16 + r + 8 * half;
        out[(size_t)m * EMB + col] = acc[i][j][r] + bb;
      }
    }
}

// ---------------------------------------------------------------------------
extern "C" void kernel_launch(void* const* d_in, const int* in_sizes, int n_in,
                              void* d_out, int out_size, void* d_ws,
                              size_t ws_size, hipStream_t stream) {
  const float* x    = (const float*)d_in[0];
  const float* Wqkv = (const float*)d_in[1];
  const float* bqkv = (const float*)d_in[2];
  const float* Wo   = (const float*)d_in[3];
  const float* bo   = (const float*)d_in[4];
  float* out = (float*)d_out;

  float* q   = (float*)d_ws;        // MROWS floats
  float* ctx = q + MROWS;           // NBP*EMB floats  (total < 1 MB of ws)

  q_gemv_kernel<<<MROWS / 8, 256, 0, stream>>>(x, Wqkv, bqkv, q);
  ctx_kernel<<<NBP, 256, 0, stream>>>(x, Wqkv, bqkv, q, ctx);
  fused_vout_kernel<<<MROWS / BM, 256, 0, stream>>>(x, Wqkv, bqkv, Wo, bo, ctx,
                                                    out);
}